// QueryAndGroupRRI_52785148068535
// MI455X (gfx1250) — compile-verified
//
#include <hip/hip_runtime.h>
#include <hip/hip_bf16.h>

typedef float v2f __attribute__((ext_vector_type(2)));
typedef float v8f __attribute__((ext_vector_type(8)));

#define RADIUS 0.1f
#define R2     (RADIUS * RADIUS)
#define NS     32      // nsample
#define NPTS   16384   // N
#define NQ     2048    // M
#define NB     4       // batch

// One wave (32 threads) per 16-query tile.
// Phase 1: ball query via V_WMMA_F32_16X16X4_F32 (d^2 as GEMM, K=4 pad),
//          register row-counters + software-pipelined point loads.
// Phase 2: fused RRI feature computation per query.
__global__ __launch_bounds__(32)
void rri_kernel(const float* __restrict__ xyz,
                const float* __restrict__ new_xyz,
                float* __restrict__ out)
{
    __shared__ float qx[16], qy[16], qz[16], qq[16];
    __shared__ int   counts[16];
    __shared__ int   idxbuf[16][NS];
    __shared__ float gpt[3][NS];      // gathered points of current query
    __shared__ float dmat[NS][NS];    // pairwise distances [i][j]
    __shared__ float redv[NS];
    __shared__ int   tip;

    const int lane = threadIdx.x;              // 0..31
    const int blk  = blockIdx.x;               // 0 .. NB*(NQ/16)-1
    const int b    = blk / (NQ / 16);
    const int q0   = (blk % (NQ / 16)) * 16;

    const float* xb = xyz     + (size_t)b * NPTS * 3;
    const float* qb = new_xyz + (size_t)b * NQ   * 3;

    // ---- load the 16 query points of this tile ----
    if (lane < 16) {
        float x = qb[(q0 + lane) * 3 + 0];
        float y = qb[(q0 + lane) * 3 + 1];
        float z = qb[(q0 + lane) * 3 + 2];
        qx[lane] = x; qy[lane] = y; qz[lane] = z;
        qq[lane] = x * x + y * y + z * z;
    }
    __syncthreads();

    const int half = lane >> 4;   // 0: lanes 0-15, 1: lanes 16-31
    const int l16  = lane & 15;
    const unsigned below = (1u << l16) - 1u;

    // A matrix 16x4 f32 (ISA 7.12.2): lane row M=l16; VGPR0 holds K=0 (lo half)/K=2 (hi),
    // VGPR1 holds K=1 (lo)/K=3 (hi).  A row = (-2x, -2y, -2z, |q|^2).
    v2f a;
    a.x = half ? (-2.0f * qz[l16]) : (-2.0f * qx[l16]);
    a.y = half ? qq[l16]           : (-2.0f * qy[l16]);

    v8f czero;
#pragma unroll
    for (int i = 0; i < 8; ++i) czero[i] = 0.0f;

    // Row counters in registers: every lane of a half-wave maintains the same
    // 8 counts (rows half*8 + k), derived from the wave-uniform ballot mask.
    int cnt[8];
#pragma unroll
    for (int k = 0; k < 8; ++k) cnt[k] = 0;

    // ---- phase 1: scan N in 16-point tiles with WMMA, 1-tile prefetch ----
    const float* p0 = xb + (size_t)l16 * 3;
    float px = p0[0], py = p0[1], pz = p0[2];

    for (int n0 = 0; n0 < NPTS; n0 += 16) {
        // prefetch next tile (uniform clamped address; wait sinks below appends)
        int nn = (n0 + 16 < NPTS) ? (n0 + 16) : n0;
        const float* pn = xb + (size_t)(nn + l16) * 3;
        float fx = pn[0], fy = pn[1], fz = pn[2];

        // ---- process current tile from registers ----
        float pr2    = px * px + py * py + pz * pz;
        float climit = R2 - pr2;                 // in-ball iff dot < climit
        v2f bb;                                  // B 4x16: column N=l16, K split by half
        bb.x = half ? pz   : px;
        bb.y = half ? 1.0f : py;

        // D[m][n] = |q_m|^2 - 2 q_m . p_n
        v8f d = __builtin_amdgcn_wmma_f32_16x16x4_f32(
            false, a, false, bb, (short)0, czero, false, false);

        // D tile layout: VGPR k -> row k (lanes 0-15) / row k+8 (lanes 16-31),
        // column = lane & 15.  Ordered append of matching indices per row.
#pragma unroll
        for (int k = 0; k < 8; ++k) {
            bool hit = d[k] < climit;
            unsigned bal = (unsigned)__ballot(hit);
            unsigned m16 = (bal >> (half * 16)) & 0xFFFFu;
            int pos = cnt[k] + __popc(m16 & below);
            if (hit && pos < NS) idxbuf[k + half * 8][pos] = n0 + l16;
            int nc = cnt[k] + __popc(m16);
            cnt[k] = nc > NS ? NS : nc;
        }

        // rotate pipeline registers
        px = fx; py = fy; pz = fz;

        // early exit once all rows are full (expected ~50% of scan at r=0.1)
        bool full = true;
#pragma unroll
        for (int k = 0; k < 8; ++k) full = full && (cnt[k] >= NS);
        if (__popcll(__ballot(full)) == 32) break;
    }

    // publish counters (lane 0 -> rows 0-7, lane 16 -> rows 8-15)
    if (l16 == 0) {
#pragma unroll
        for (int k = 0; k < 8; ++k) counts[k + half * 8] = cnt[k];
    }
    __syncthreads();

    // pad short lists: fill with first match, or 0 if no match (matches reference)
    if (lane < 16) {
        int c     = counts[lane];
        int first = (c > 0) ? idxbuf[lane][0] : 0;
        for (int p2 = c; p2 < NS; ++p2) idxbuf[lane][p2] = first;
    }
    __syncthreads();

    // ---- phase 2: per-query RRI features; lane j owns sample j ----
    for (int q = 0; q < 16; ++q) {
        const int m = q0 + q;
        float cx = qx[q], cy = qy[q], czv = qz[q];
        float cc = qq[q];

        int gi = idxbuf[q][lane];
        const float* gp = xb + (size_t)gi * 3;
        float gxv = gp[0], gyv = gp[1], gzv = gp[2];
        gpt[0][lane] = gxv; gpt[1][lane] = gyv; gpt[2][lane] = gzv;
        float gr = sqrtf(gxv * gxv + gyv * gyv + gzv * gzv);
        __syncthreads();

        // column j of pairwise distance matrix; symmetric => row mean == col sum
        float colsum = 0.0f;
        for (int i = 0; i < NS; ++i) {
            float dx = gpt[0][i] - gxv;
            float dy = gpt[1][i] - gyv;
            float dz = gpt[2][i] - gzv;
            float dd = sqrtf(dx * dx + dy * dy + dz * dz);
            dmat[i][lane] = dd;
            colsum += dd;
        }
        redv[lane] = colsum;
        __syncthreads();
        if (lane == 0) {                 // argmax, first occurrence wins
            float bv = redv[0]; int bi = 0;
            for (int i = 1; i < NS; ++i)
                if (redv[i] > bv) { bv = redv[i]; bi = i; }
            tip = bi;
        }
        __syncthreads();
        float tx = gpt[0][tip], ty = gpt[1][tip], tz = gpt[2][tip];

        // gpv = (c x g) x c = g*(c.c) - c*(c.g), normalized
        float dotcg = cx * gxv + cy * gyv + czv * gzv;
        float gpx = gxv * cc - cx * dotcg;
        float gpy = gyv * cc - cy * dotcg;
        float gpz = gzv * cc - czv * dotcg;
        float gn  = sqrtf(gpx * gpx + gpy * gpy + gpz * gpz);
        gpx /= gn; gpy /= gn; gpz /= gn;

        // tpv = (c x tip) x c, normalized
        float dotct = cx * tx + cy * ty + czv * tz;
        float tpx = tx * cc - cx * dotct;
        float tpy = ty * cc - cy * dotct;
        float tpz = tz * cc - czv * dotct;
        float tn  = sqrtf(tpx * tpx + tpy * tpy + tpz * tpz);
        tpx /= tn; tpy /= tn; tpz /= tn;

        // gcv = gpv x tpv ; sinv = gcv . (c / (|c|+1e-8))
        float gcx = gpy * tpz - gpz * tpy;
        float gcy = gpz * tpx - gpx * tpz;
        float gcz = gpx * tpy - gpy * tpx;
        float inr = 1.0f / (sqrtf(cc) + 1e-8f);
        float sinv = (gcx * cx + gcy * cy + gcz * czv) * inr;

        // ascending insertion sort of column j in LDS (conflict-free banks)
        for (int i = 1; i < NS; ++i) {
            float key = dmat[i][lane];
            int w = i - 1;
            while (w >= 0 && dmat[w][lane] > key) {
                dmat[w + 1][lane] = dmat[w][lane];
                --w;
            }
            dmat[w + 1][lane] = key;
        }

        // output (B, 33, M, 32): 32 sorted*sinv channels + grouped radius
        size_t base = (((size_t)b * 33) * NQ + m) * NS + lane;
        const size_t chs = (size_t)NQ * NS;
        for (int i = 0; i < NS; ++i)
            out[base + (size_t)i * chs] = dmat[i][lane] * sinv;
        out[base + (size_t)NS * chs] = gr;
        __syncthreads();   // protect gpt/dmat before next query
    }
}

extern "C" void kernel_launch(void* const* d_in, const int* in_sizes, int n_in,
                              void* d_out, int out_size, void* d_ws, size_t ws_size,
                              hipStream_t stream) {
    const float* xyz     = (const float*)d_in[0];
    const float* new_xyz = (const float*)d_in[1];
    float* out = (float*)d_out;
    dim3 grid(NB * (NQ / 16));
    dim3 block(32);
    hipLaunchKernelGGL(rri_kernel, grid, block, 0, stream, xyz, new_xyz, out);
}